// RandLANet_86852828660098
// MI455X (gfx1250) — compile-verified
//
#include <hip/hip_runtime.h>

#define EPSF 1e-6f

typedef __attribute__((ext_vector_type(2))) float v2f;
typedef __attribute__((ext_vector_type(8))) float v8f;

__device__ __forceinline__ v8f wmma4(v2f a, v2f b, v8f c)
{
    return __builtin_amdgcn_wmma_f32_16x16x4_f32(false, a, false, b, (short)0, c, false, false);
}

// ---------------------------------------------------------------------------
// Generic fused conv1x1 GEMM:  out[b,o,p] = act( bn( W[o,:]·in[b,:,p] + bias[o] ) + res[b,o,p] )
// One wave per 16x16 output tile, K stepped by 4 via V_WMMA_F32_16X16X4_F32.
// Layouts per CDNA5 ISA 7.12.2:
//   A (16x4, f32): lane l<16 -> row M=l, K={0,1} in v0,v1 ; lane l>=16 -> K={2,3}
//   B (4x16, f32): v_j lanes 0-15 -> K=j, N=lane ; lanes 16-31 -> K=j+2
//   C/D (16x16):   v_j lanes 0-15 -> M=j ; lanes 16-31 -> M=j+8, N=lane&15
// Interior tiles take a branch-free unguarded path (uniform scalar branch);
// ragged tiles use clamped-index unconditional loads + v_cndmask selects so
// EXEC is never modified around the WMMA.
// ---------------------------------------------------------------------------
__global__ __launch_bounds__(32) void conv1x1_wmma_k(
    const float* __restrict__ W, const float* __restrict__ bias,
    const float* __restrict__ g, const float* __restrict__ be,
    const float* __restrict__ mn, const float* __restrict__ vr,
    const float* __restrict__ in, const float* __restrict__ res,
    float* __restrict__ out, int O, int C, int P, int Ototal, int act)
{
    const int b    = blockIdx.z;
    const int o0   = blockIdx.y << 4;
    const int p0   = blockIdx.x << 4;
    const int lane = threadIdx.x & 31;
    const int half = lane >> 4;
    const int l15  = lane & 15;
    const float* inb = in + (size_t)b * C * P;

    const int  ar   = o0 + l15;
    const int  pc   = p0 + l15;
    const bool arok = (ar < O);
    const bool pok  = (pc < P);
    const float* Wp = W + (size_t)(arok ? ar : (O - 1)) * C;   // clamped row
    const float* Bp = inb + (pok ? pc : (P - 1));              // clamped col

    v8f acc = {};
    const int Cm = C & ~3;
    int c0 = 0;

    if ((o0 + 16 <= O) && (p0 + 16 <= P)) {
        // -------- interior tile: unguarded loads, unrolled x2 --------
        for (; c0 + 8 <= Cm; c0 += 8) {
            const int ac0 = c0 + 2 * half;
            const int ac1 = ac0 + 4;
            v2f a0, b0, a1, b1;
            a0.x = Wp[ac0];                 a0.y = Wp[ac0 + 1];
            b0.x = Bp[(size_t)ac0 * P];     b0.y = Bp[(size_t)(ac0 + 1) * P];
            a1.x = Wp[ac1];                 a1.y = Wp[ac1 + 1];
            b1.x = Bp[(size_t)ac1 * P];     b1.y = Bp[(size_t)(ac1 + 1) * P];
            acc = wmma4(a0, b0, acc);
            acc = wmma4(a1, b1, acc);
        }
        for (; c0 < Cm; c0 += 4) {
            const int ac = c0 + 2 * half;
            v2f a, bv;
            a.x  = Wp[ac];                  a.y  = Wp[ac + 1];
            bv.x = Bp[(size_t)ac * P];      bv.y = Bp[(size_t)(ac + 1) * P];
            acc = wmma4(a, bv, acc);
        }
    }
    // -------- guarded loop: whole K for ragged tiles, C%4 tail for interior --------
    for (; c0 < C; c0 += 4) {
        const int ac  = c0 + 2 * half;
        const int i0  = (ac     < C) ? ac     : (C - 1);
        const int i1  = (ac + 1 < C) ? ac + 1 : (C - 1);
        const float wx = Wp[i0],                 wy = Wp[i1];
        const float bx = Bp[(size_t)i0 * P],     by = Bp[(size_t)i1 * P];
        v2f a, bv;
        a.x  = (arok && ac     < C) ? wx : 0.0f;
        a.y  = (arok && ac + 1 < C) ? wy : 0.0f;
        bv.x = (pok  && ac     < C) ? bx : 0.0f;
        bv.y = (pok  && ac + 1 < C) ? by : 0.0f;
        acc = wmma4(a, bv, acc);
    }

    if (pok) {
#pragma unroll
        for (int j = 0; j < 8; ++j) {
            const int o = o0 + j + 8 * half;
            if (o < O) {
                float v = acc[j];
                if (bias) v += bias[o];
                if (g) {
                    float s = g[o] * rsqrtf(vr[o] + EPSF);
                    v = (v - mn[o]) * s + be[o];
                }
                if (res) v += res[((size_t)b * Ototal + o) * P + pc];
                if      (act == 1) v = v > 0.0f ? v : 0.0f;
                else if (act == 2) v = v > 0.0f ? v : 0.2f  * v;
                else if (act == 3) v = v > 0.0f ? v : 0.01f * v;
                out[((size_t)b * Ototal + o) * P + pc] = v;
            }
        }
    }
}

// ---------------------------------------------------------------------------
// KNN: per-query top-KN smallest d2 = |q|^2 + |s|^2 - 2 q·s, LDS-tiled support,
// with global_prefetch of the next tile. Strict '<' comparisons reproduce
// lax.top_k tie-breaking (earlier index wins).
// ---------------------------------------------------------------------------
template <int KN>
__global__ __launch_bounds__(256) void knn_k(
    const float* __restrict__ coords, int N, int ns, int nq,
    int* __restrict__ idx, float* __restrict__ dist)
{
    __shared__ float sx[256], sy[256], sz[256], ssq[256];
    const int tid = threadIdx.x;
    const int q   = blockIdx.x * 256 + tid;
    const int b   = blockIdx.y;
    const float* cb = coords + (size_t)b * N * 3;

    float qx = 0.f, qy = 0.f, qz = 0.f;
    if (q < nq) { qx = cb[q * 3]; qy = cb[q * 3 + 1]; qz = cb[q * 3 + 2]; }
    const float qq = qx * qx + qy * qy + qz * qz;

    float bestd[KN]; int besti[KN];
#pragma unroll
    for (int k = 0; k < KN; ++k) { bestd[k] = 3.0e38f; besti[k] = 0; }

    for (int s0 = 0; s0 < ns; s0 += 256) {
        __syncthreads();
        const int sl = s0 + tid;
        if (sl < ns) {
            float x = cb[sl * 3], y = cb[sl * 3 + 1], z = cb[sl * 3 + 2];
            sx[tid] = x; sy[tid] = y; sz[tid] = z; ssq[tid] = x * x + y * y + z * z;
        }
        __syncthreads();
        const int nxt = s0 + 256 + tid;              // prefetch next tile into cache
        if (nxt < ns) __builtin_prefetch(cb + nxt * 3, 0, 3);
        const int tl = min(256, ns - s0);
        for (int j = 0; j < tl; ++j) {
            float d2 = qq + ssq[j] - 2.0f * (qx * sx[j] + qy * sy[j] + qz * sz[j]);
            if (d2 < bestd[KN - 1]) {
                bestd[KN - 1] = d2; besti[KN - 1] = s0 + j;
#pragma unroll
                for (int k = KN - 1; k > 0; --k) {
                    if (bestd[k] < bestd[k - 1]) {
                        float td = bestd[k]; bestd[k] = bestd[k - 1]; bestd[k - 1] = td;
                        int   ti = besti[k]; besti[k] = besti[k - 1]; besti[k - 1] = ti;
                    }
                }
            }
        }
    }

    if (q < nq) {
#pragma unroll
        for (int k = 0; k < KN; ++k) {
            idx[((size_t)b * nq + q) * KN + k] = besti[k];
            if (dist)
                dist[((size_t)b * nq + q) * KN + k] = sqrtf(fmaxf(bestd[k], 1e-12f));
        }
    }
}

// coords[b,n,0:3] = x[b,n,0:3]
__global__ void extract_coords_k(const float* __restrict__ x, float* __restrict__ c, int total)
{
    int t = blockIdx.x * blockDim.x + threadIdx.x;
    if (t >= total) return;
    c[t * 3 + 0] = x[t * 6 + 0];
    c[t * 3 + 1] = x[t * 6 + 1];
    c[t * 3 + 2] = x[t * 6 + 2];
}

// fc_start (6->8) + bn_start + leaky_relu(0.2), output (B,8,N) channel-major.
__global__ void fc_start_k(const float* __restrict__ x, const float* __restrict__ W,
                           const float* __restrict__ bias,
                           const float* __restrict__ g, const float* __restrict__ be,
                           const float* __restrict__ mn, const float* __restrict__ vr,
                           float* __restrict__ out, int N, int total)
{
    int t = blockIdx.x * blockDim.x + threadIdx.x;
    if (t >= total) return;
    int b = t / (8 * N);
    int r = t - b * 8 * N;
    int o = r / N;
    int n = r - o * N;
    const float* xr = x + ((size_t)b * N + n) * 6;
    float acc = bias[o];
#pragma unroll
    for (int d = 0; d < 6; ++d) acc += xr[d] * W[o * 6 + d];
    float s = g[o] * rsqrtf(vr[o] + EPSF);
    acc = (acc - mn[o]) * s + be[o];
    out[t] = acc > 0.0f ? acc : 0.2f * acc;
}

// rel = concat([center(3), neigh(3), center-neigh(3), dist(1)]) -> (B,10,n,16)
__global__ void build_rel_k(const float* __restrict__ coords, const int* __restrict__ idx,
                            const float* __restrict__ dist, float* __restrict__ rel,
                            int n, int N, int total)
{
    int t = blockIdx.x * blockDim.x + threadIdx.x;
    if (t >= total) return;
    int k = t & 15;
    int r = t >> 4;
    int q = r % n;
    int b = r / n;
    const float* cb = coords + (size_t)b * N * 3;
    int id = idx[((size_t)b * n + q) * 16 + k];
    float cx = cb[q * 3], cy = cb[q * 3 + 1], cz = cb[q * 3 + 2];
    float nx = cb[id * 3], ny = cb[id * 3 + 1], nz = cb[id * 3 + 2];
    size_t base = (((size_t)b * 10) * n + q) * 16 + k;
    size_t st = (size_t)n * 16;
    rel[base + 0 * st] = cx;       rel[base + 1 * st] = cy;       rel[base + 2 * st] = cz;
    rel[base + 3 * st] = nx;       rel[base + 4 * st] = ny;       rel[base + 5 * st] = nz;
    rel[base + 6 * st] = cx - nx;  rel[base + 7 * st] = cy - ny;  rel[base + 8 * st] = cz - nz;
    rel[base + 9 * st] = dist[((size_t)b * n + q) * 16 + k];
}

// lse feature broadcast: out channels [Hf, 2*Hf) of Do-channel buffer.
__global__ void bcast_k(const float* __restrict__ xh, float* __restrict__ lse,
                        int Hf, int Do, int n, int total)
{
    int t = blockIdx.x * blockDim.x + threadIdx.x;
    if (t >= total) return;
    int k = t & 15;
    int r = t >> 4;
    int q = r % n;  r /= n;
    int c = r % Hf;
    int b = r / Hf;
    lse[(((size_t)b * Do + Hf + c) * n + q) * 16 + k] =
        xh[((size_t)b * Hf + c) * n + q];
}

// attention pool: softmax over K=16 of scores, weighted sum of x -> (B,C,n)
__global__ void softmax_pool_k(const float* __restrict__ x, const float* __restrict__ s,
                               float* __restrict__ out, int C, int n, int total)
{
    int t = blockIdx.x * blockDim.x + threadIdx.x;
    if (t >= total) return;
    int q = t % n;
    int r = t / n;
    int c = r % C;
    int b = r / C;
    size_t base = (((size_t)b * C + c) * n + q) * 16;
    float m = -3.0e38f;
#pragma unroll
    for (int k = 0; k < 16; ++k) m = fmaxf(m, s[base + k]);
    float Z = 0.0f, acc = 0.0f;
#pragma unroll
    for (int k = 0; k < 16; ++k) {
        float e = expf(s[base + k] - m);
        Z += e;
        acc += e * x[base + k];
    }
    out[((size_t)b * C + c) * n + q] = acc / Z;
}

// h[:, :, :np2]   (channel-major slice-copy)
__global__ void downsample_k(const float* __restrict__ in, float* __restrict__ out,
                             int C, int np, int np2, int total)
{
    int t = blockIdx.x * blockDim.x + threadIdx.x;
    if (t >= total) return;
    int p = t % np2;
    int r = t / np2;
    int c = r % C;
    int b = r / C;
    out[((size_t)b * C + c) * np2 + p] = in[((size_t)b * C + c) * np + p];
}

// decoder 1-NN gather into channels [0,C) of a 2C-channel concat buffer
__global__ void gather_k(const float* __restrict__ h, const int* __restrict__ idx1,
                         float* __restrict__ dst, int C, int ns, int nq, int total)
{
    int t = blockIdx.x * blockDim.x + threadIdx.x;
    if (t >= total) return;
    int q = t % nq;
    int r = t / nq;
    int c = r % C;
    int b = r / C;
    int id = idx1[(size_t)b * nq + q];
    dst[((size_t)b * (2 * C) + c) * nq + q] = h[((size_t)b * C + c) * ns + id];
}

// skip-connection copy into channels [C,2C)
__global__ void concat_k(const float* __restrict__ src, float* __restrict__ dst,
                         int C, int nq, int total)
{
    int t = blockIdx.x * blockDim.x + threadIdx.x;
    if (t >= total) return;
    int q = t % nq;
    int r = t / nq;
    int c = r % C;
    int b = r / C;
    dst[((size_t)b * (2 * C) + C + c) * nq + q] = src[((size_t)b * C + c) * nq + q];
}

// ---------------------------------------------------------------------------
extern "C" void kernel_launch(void* const* d_in, const int* in_sizes, int n_in,
                              void* d_out, int out_size, void* d_ws, size_t ws_size,
                              hipStream_t stream)
{
    (void)in_sizes; (void)n_in; (void)out_size; (void)ws_size;
    const int B = 2, N = 8192;

    auto F = [&](int i) { return (const float*)d_in[i]; };

    // ---- flattened-pytree parameter indices (x first, then jax leaf order) ----
    // top keys sorted: bn_start, dec, enc, fc_end, fc_start, mid
    // bn: beta,gamma,mean,var ; smlp: W,b[,bn...] ; pool: mlp(W,b,bn),score_W
    const int IBN = 1;                               // bn_start beta,gamma,mean,var
    auto decI = [&](int i) { return 5 + 6 * i; };    // dec[i]: W,b,beta,gamma,mean,var
    auto lfaI = [&](int l) { return 29 + 36 * l; };  // per-lfa base
    // lfa offsets: lse1 +0..5 | lse2 +6..11 | mlp1 +12,13 | mlp2 +14,15
    //              pool1.mlp +16..21 | pool1.score_W +22
    //              pool2.mlp +23..28 | pool2.score_W +29 | shortcut +30..35
    const int IFE = 173;   // fc_end[0] 173..178, [1] 179..184, [2] W=185,b=186
    const int IFS = 187;   // fc_start W,b
    const int IMID = 189;  // mid W,b

    // ---- workspace bump allocator ----
    char* wsp = (char*)d_ws; size_t off = 0;
    auto alloc  = [&](size_t e) { float* p = (float*)(wsp + off); off += ((e * 4 + 255) & ~(size_t)255); return p; };
    auto alloci = [&](size_t e) { int*   p = (int*)  (wsp + off); off += ((e * 4 + 255) & ~(size_t)255); return p; };

    float* coords = alloc((size_t)B * N * 3);
    float* h0     = alloc((size_t)B * 8 * N);
    const int stkC[4] = {32, 128, 256, 512};
    const int stkN[4] = {8192, 2048, 512, 128};
    float* stk[4];
    for (int i = 0; i < 4; ++i) stk[i] = alloc((size_t)B * stkC[i] * stkN[i]);
    float* hds   = alloc((size_t)B * 65536);
    float* hmid  = alloc((size_t)B * 512 * 32);
    int*   idxb  = alloci((size_t)B * 8192 * 16);
    float* distb = alloc((size_t)B * 8192 * 16);
    float* xh    = alloc((size_t)B * 65536);
    float* rel   = alloc((size_t)B * 10 * 8192 * 16);
    float* lseb  = alloc((size_t)B * 2097152);
    float* scb   = alloc((size_t)B * 2097152);
    float* poolA = alloc((size_t)B * 131072);
    float* poolB = alloc((size_t)B * 131072);
    float* shb   = alloc((size_t)B * 262144);
    int*   idx1  = alloci((size_t)B * 8192);
    float* dcat  = alloc((size_t)B * 524288);
    float* dA    = alloc((size_t)B * 65536);
    float* dB    = alloc((size_t)B * 65536);
    float* f0    = alloc((size_t)B * 64 * 8192);
    float* f1    = alloc((size_t)B * 32 * 8192);

    auto conv = [&](const float* W, const float* bias, const float* g, const float* be,
                    const float* mn, const float* vr, const float* in, const float* res,
                    float* out, int O, int C, int P, int Ototal, int act) {
        dim3 gr((P + 15) / 16, (O + 15) / 16, B);
        conv1x1_wmma_k<<<gr, 32, 0, stream>>>(W, bias, g, be, mn, vr, in, res, out,
                                              O, C, P, Ototal, act);
    };
    // smlp by flat index: W=ib, b=ib+1, [beta=ib+2, gamma=ib+3, mean=ib+4, var=ib+5]
    auto convS = [&](int ib, bool bn, const float* in, const float* res, float* out,
                     int O, int C, int P, int Ototal, int act) {
        conv(F(ib), F(ib + 1),
             bn ? F(ib + 3) : nullptr, bn ? F(ib + 2) : nullptr,
             bn ? F(ib + 4) : nullptr, bn ? F(ib + 5) : nullptr,
             in, res, out, O, C, P, Ototal, act);
    };
    auto g1 = [](int total) { return dim3((total + 255) / 256); };

    // ---- stem ----
    extract_coords_k<<<g1(B * N), 256, 0, stream>>>(F(0), coords, B * N);
    fc_start_k<<<g1(B * N * 8), 256, 0, stream>>>(F(0), F(IFS), F(IFS + 1),
        F(IBN + 1), F(IBN + 0), F(IBN + 2), F(IBN + 3), h0, N, B * N * 8);

    // ---- encoder ----
    const int dinL[4]  = {8, 32, 128, 256};
    const int doutL[4] = {16, 64, 128, 256};
    float* feat = h0;
    int np = N;
    for (int l = 0; l < 4; ++l) {
        const int Cf = dinL[l], Do = doutL[l], Hf = Do / 2, Co = 2 * Do;
        const int L = lfaI(l);
        knn_k<16><<<dim3((np + 255) / 256, B), 256, 0, stream>>>(coords, N, np, np, idxb, distb);
        // mlp1: Cf -> Hf, lrelu(0.2)
        conv(F(L + 12), F(L + 13), nullptr, nullptr, nullptr, nullptr,
             feat, nullptr, xh, Hf, Cf, np, Hf, 2);
        // lse1
        build_rel_k<<<g1(B * np * 16), 256, 0, stream>>>(coords, idxb, distb, rel, np, N, B * np * 16);
        convS(L + 0, true, rel, nullptr, lseb, Hf, 10, np * 16, Do, 1);
        bcast_k<<<g1(B * Hf * np * 16), 256, 0, stream>>>(xh, lseb, Hf, Do, np, B * Hf * np * 16);
        // pool1
        conv(F(L + 22), nullptr, nullptr, nullptr, nullptr, nullptr,
             lseb, nullptr, scb, Do, Do, np * 16, Do, 0);
        softmax_pool_k<<<g1(B * Do * np), 256, 0, stream>>>(lseb, scb, poolA, Do, np, B * Do * np);
        convS(L + 16, true, poolA, nullptr, xh, Hf, Do, np, Hf, 1);
        // lse2 (same idx/dist/rel)
        convS(L + 6, true, rel, nullptr, lseb, Hf, 10, np * 16, Do, 1);
        bcast_k<<<g1(B * Hf * np * 16), 256, 0, stream>>>(xh, lseb, Hf, Do, np, B * Hf * np * 16);
        // pool2
        conv(F(L + 29), nullptr, nullptr, nullptr, nullptr, nullptr,
             lseb, nullptr, scb, Do, Do, np * 16, Do, 0);
        softmax_pool_k<<<g1(B * Do * np), 256, 0, stream>>>(lseb, scb, poolA, Do, np, B * Do * np);
        convS(L + 23, true, poolA, nullptr, poolB, Do, Do, np, Do, 1);
        // shortcut (bn, no act), then mlp2 + residual + lrelu(0.01)
        convS(L + 30, true, feat, nullptr, shb, Co, Cf, np, Co, 0);
        conv(F(L + 14), F(L + 15), nullptr, nullptr, nullptr, nullptr,
             poolB, shb, stk[l], Co, Do, np, Co, 3);
        // downsample x4
        const int np2 = np / 4;
        downsample_k<<<g1(B * Co * np2), 256, 0, stream>>>(stk[l], hds, Co, np, np2, B * Co * np2);
        feat = hds; np = np2;
    }

    // ---- mid: 512 -> 512, relu ----
    conv(F(IMID), F(IMID + 1), nullptr, nullptr, nullptr, nullptr,
         feat, nullptr, hmid, 512, 512, np, 512, 1);

    // ---- decoder ----
    const int decO[4] = {256, 128, 32, 8};
    float* h = hmid; int Ch = 512; int ns = 32;
    float* pp[2] = {dA, dB};
    for (int i = 0; i < 4; ++i) {
        const int nq = ns * 4;
        knn_k<1><<<dim3((nq + 255) / 256, B), 256, 0, stream>>>(coords, N, ns, nq, idx1, nullptr);
        gather_k<<<g1(B * Ch * nq), 256, 0, stream>>>(h, idx1, dcat, Ch, ns, nq, B * Ch * nq);
        concat_k<<<g1(B * Ch * nq), 256, 0, stream>>>(stk[3 - i], dcat, Ch, nq, B * Ch * nq);
        convS(decI(i), true, dcat, nullptr, pp[i & 1], decO[i], 2 * Ch, nq, decO[i], 1);
        h = pp[i & 1]; Ch = decO[i]; ns = nq;
    }

    // ---- fc_end: 8->64 (bn,relu), 64->32 (bn,relu), 32->13 ----
    convS(IFE + 0, true, h,  nullptr, f0, 64, 8,  8192, 64, 1);
    convS(IFE + 6, true, f0, nullptr, f1, 32, 64, 8192, 32, 1);
    conv(F(IFE + 12), F(IFE + 13), nullptr, nullptr, nullptr, nullptr,
         f1, nullptr, (float*)d_out, 13, 32, 8192, 13, 0);
}